// NeuralGraphPool_39049842655951
// MI455X (gfx1250) — compile-verified
//
#include <hip/hip_runtime.h>

// NeuralGraphPool for MI455X (gfx1250, wave32).
// out[b,a,:] = max(atoms[b,a,:], max over valid neighbors atoms[b,e,:]) * (deg!=0)
// Memory-bound gather+max: ~129 MB HBM traffic -> ~5.5us roofline @ 23.3 TB/s.
// One wave per row; float4 per lane => 512B per vmem op. Gather amplification
// (9x, 576 MB) served from L2 (atoms = 64 MB << 192 MB L2).

typedef float v4f __attribute__((ext_vector_type(4)));

namespace {
constexpr int kB = 256;
constexpr int kA = 128;
constexpr int kD = 8;
constexpr int kF = 512;            // floats per row
constexpr int kWavesPerBlock = 8;  // 256 threads = 8 wave32
constexpr int kRows = kB * kA;     // 32768
}

__global__ __launch_bounds__(kWavesPerBlock * 32, 4)
void ngp_gather_max_kernel(const float* __restrict__ atoms,
                           const int*   __restrict__ edges,
                           float*       __restrict__ out) {
  const int lane = threadIdx.x & 31;
  const int wave = threadIdx.x >> 5;
  const int row  = blockIdx.x * kWavesPerBlock + wave;   // [0, kRows)

  const int    b         = row >> 7;                     // row / kA (kA == 128)
  const size_t rowBase   = (size_t)row * kF;
  const size_t batchBase = (size_t)b * (size_t)kA * kF;

  // ---- Self-row loads: unconditional, issue alongside the edge loads ------
  const v4f* __restrict__ sp = (const v4f*)(atoms + rowBase);
  v4f acc[4];
#pragma unroll
  for (int p = 0; p < 4; ++p)
    acc[p] = sp[p * 32 + lane];

  const int* __restrict__ ep = edges + (size_t)row * kD;

  // ---- Early per-lane prefetch of neighbor rows (global_prefetch_b8) ------
  // Per-lane edge value is available before the readfirstlane->SALU chain
  // resolves; fan 32 lanes over 8 neighbors x 4 positions (512B apart) and
  // pull 2 cachelines each, overlapping gather latency with scalarization.
  {
    const int eLane    = ep[lane & 7];
    const int eClamped = (eLane < 0) ? 0 : eLane;  // stay in-bounds (speculative)
    const char* nbase  =
        (const char*)(atoms + batchBase + (size_t)eClamped * kF);
    const int pos = (lane >> 3) * 512;
    __builtin_prefetch(nbase + pos, 0, 3);
    __builtin_prefetch(nbase + pos + 128, 0, 3);
  }

  // ---- Edge indices are wave-uniform: pull into SGPRs -> scalar control ---
  int e[kD];
  int deg = 0;
#pragma unroll
  for (int d = 0; d < kD; ++d) {
    e[d] = __builtin_amdgcn_readfirstlane(ep[d]);   // uniform, scalar branches
    deg += (e[d] >= 0) ? 1 : 0;
  }

  // ---- Max-accumulate valid neighbors (skip invalid: no -inf pad loads) ---
#pragma unroll
  for (int d = 0; d < kD; ++d) {
    if (e[d] >= 0) {
      const v4f* __restrict__ np =
          (const v4f*)(atoms + batchBase + (size_t)e[d] * kF);
#pragma unroll
      for (int p = 0; p < 4; ++p) {
        v4f n = np[p * 32 + lane];
#pragma unroll
        for (int c = 0; c < 4; ++c)
          acc[p][c] = fmaxf(acc[p][c], n[c]);
      }
    }
  }

  // ---- Degree mask as a uniform multiply (matches reference exactly; ------
  //      removes the branch so self loads stay unconditional/hoistable)
  const float mask = (deg != 0) ? 1.0f : 0.0f;

  // ---- Streamed output: non-temporal (never re-read; keep L2 for atoms) ---
  v4f* __restrict__ o4 = (v4f*)(out + rowBase);
#pragma unroll
  for (int p = 0; p < 4; ++p) {
    v4f r = acc[p] * mask;
    __builtin_nontemporal_store(r, &o4[p * 32 + lane]);
  }
}

extern "C" void kernel_launch(void* const* d_in, const int* in_sizes, int n_in,
                              void* d_out, int out_size, void* d_ws, size_t ws_size,
                              hipStream_t stream) {
  // Inputs (setup_inputs order): 0=atoms f32[B,A,F], 1=bonds (UNUSED), 2=edges i32[B,A,D]
  const float* atoms = (const float*)d_in[0];
  const int*   edges = (const int*)d_in[2];
  float*       out   = (float*)d_out;     // f32[B,A,F]

  (void)in_sizes; (void)n_in; (void)out_size; (void)d_ws; (void)ws_size;

  const dim3 grid(kRows / kWavesPerBlock);   // 4096 blocks
  const dim3 block(kWavesPerBlock * 32);     // 256 threads = 8 wave32
  ngp_gather_max_kernel<<<grid, block, 0, stream>>>(atoms, edges, out);
}